// AdditiveAttention_3040836845968
// MI455X (gfx1250) — compile-verified
//
#include <hip/hip_runtime.h>
#include <math.h>

#define BATCH 4
#define NQ 128
#define NK 1024
#define DQ 256
#define DK 256
#define HH 256
#define DV 256
#define KT 32  // k-rows per score block

typedef float v2f __attribute__((ext_vector_type(2)));
typedef float v8f __attribute__((ext_vector_type(8)));

__device__ __forceinline__ float fast_tanh(float x) {
#if __has_builtin(__builtin_amdgcn_tanhf)
  return __builtin_amdgcn_tanhf(x);   // v_tanh_f32 (gfx1250 TRANS op)
#else
  return tanhf(x);
#endif
}

// Async global->LDS copy (CDNA5, ASYNCcnt-tracked, bypasses VGPRs).
// vdst VGPR holds the wave-relative LDS byte address = low 32 bits of the
// flat pointer to a __shared__ object.
__device__ __forceinline__ void async_copy_b128(void* lds_dst, const float* gsrc) {
  unsigned lds_off = (unsigned)(unsigned long long)lds_dst;
  asm volatile("global_load_async_to_lds_b128 %0, %1, off"
               :: "v"(lds_off), "v"(gsrc) : "memory");
}
__device__ __forceinline__ void wait_async_lds() {
  asm volatile("s_wait_asynccnt 0x0" ::: "memory");
}

// ---------------------------------------------------------------------------
// Batched row-major f32 GEMM  C = A * B  using V_WMMA_F32_16X16X4_F32.
// One wave32 owns a 32x32 supertile (4 x 16x16 accumulators); K steps by 4.
// Per K-step: 2 A-fragments (contiguous float2 each) + 2 B-fragments
// (2 strided b32 each) feed 4 WMMAs -> 4x arithmetic density vs 1 tile/wave.
// Fragment layouts per ISA 7.12.2:
//   A: lane r(0..15)/half h -> A[row, k0+2h .. k0+2h+1]
//   B: b.x = B[k0+2h, col], b.y = B[k0+2h+1, col]
//   C/D: VGPR vr -> row base + vr + 8*half, col base + r
// ---------------------------------------------------------------------------
__global__ void wmma_gemm_f32(const float* __restrict__ A,
                              const float* __restrict__ Bm,
                              float* __restrict__ C,
                              int M, int N, int K,
                              int lda, int ldb, int ldc,
                              long long sA, long long sB, long long sC,
                              int totalTiles)
{
  const int wave   = threadIdx.x >> 5;
  const int lane   = threadIdx.x & 31;
  const int tileId = blockIdx.x * (blockDim.x >> 5) + wave;
  if (tileId >= totalTiles) return;          // wave-uniform: EXEC stays all-1s

  const int tilesN2 = N >> 5;                // 32-wide supertiles
  const int tilesM2 = M >> 5;
  const int perBat  = tilesM2 * tilesN2;
  const int batch   = tileId / perBat;
  const int rem     = tileId - batch * perBat;
  const int tm      = rem / tilesN2;
  const int tn      = rem - tm * tilesN2;

  const int r    = lane & 15;
  const int half = lane >> 4;
  const int koff = half << 1;

  const float* Ab0 = A  + (long long)batch * sA + (long long)(tm * 32 + r)      * lda;
  const float* Ab1 = A  + (long long)batch * sA + (long long)(tm * 32 + 16 + r) * lda;
  const float* Bb0 = Bm + (long long)batch * sB + (tn * 32 + r);
  const float* Bb1 = Bm + (long long)batch * sB + (tn * 32 + 16 + r);

  v8f acc00 = {}, acc01 = {}, acc10 = {}, acc11 = {};
  for (int k0 = 0; k0 < K; k0 += 4) {
    const long long kr0 = (long long)(k0 + koff) * ldb;
    const long long kr1 = kr0 + ldb;
    v2f a0 = *(const v2f*)(Ab0 + k0 + koff);
    v2f a1 = *(const v2f*)(Ab1 + k0 + koff);
    v2f b0, b1;
    b0.x = Bb0[kr0];  b0.y = Bb0[kr1];
    b1.x = Bb1[kr0];  b1.y = Bb1[kr1];
    acc00 = __builtin_amdgcn_wmma_f32_16x16x4_f32(false, a0, false, b0, (short)0, acc00, false, false);
    acc01 = __builtin_amdgcn_wmma_f32_16x16x4_f32(false, a0, false, b1, (short)0, acc01, false, false);
    acc10 = __builtin_amdgcn_wmma_f32_16x16x4_f32(false, a1, false, b0, (short)0, acc10, false, false);
    acc11 = __builtin_amdgcn_wmma_f32_16x16x4_f32(false, a1, false, b1, (short)0, acc11, false, false);
  }

  float* Crow0 = C + (long long)batch * sC + (long long)(tm * 32)      * ldc + tn * 32 + r;
  float* Crow1 = C + (long long)batch * sC + (long long)(tm * 32 + 16) * ldc + tn * 32 + r;
#pragma unroll
  for (int vr = 0; vr < 8; ++vr) {
    const long long ro = (long long)(vr + 8 * half) * ldc;
    Crow0[ro]      = acc00[vr];
    Crow0[ro + 16] = acc01[vr];
    Crow1[ro]      = acc10[vr];
    Crow1[ro + 16] = acc11[vr];
  }
}

// ---------------------------------------------------------------------------
// scores[b,q,k] = sum_h v_w[h] * tanh(qh[b,q,h] + kh[b,k,h])
// Block = 256 threads, tile = 16 q-rows x 32 k-rows. Tiles staged into LDS
// with async b128 copies (ASYNCcnt path); k rows padded to stride 257 floats
// so 32 distinct k-rows hit 32 distinct banks (257 % 64 == 1).
// ---------------------------------------------------------------------------
__global__ void additive_scores(const float* __restrict__ qh,
                                const float* __restrict__ kh,
                                const float* __restrict__ v_w,
                                float* __restrict__ scores)
{
  __shared__ float qs[16 * HH];
  __shared__ float ks[KT * (HH + 1)];
  __shared__ float vs[HH];

  const int kTiles = NK / KT;                 // 32
  const int qTiles = NQ / 16;                 // 8
  const int kt = blockIdx.x % kTiles;
  const int qt = (blockIdx.x / kTiles) % qTiles;
  const int b  = blockIdx.x / (kTiles * qTiles);

  const int tid = threadIdx.x;
  const float* qbase = qh + ((long long)b * NQ + qt * 16) * HH;
  const float* kbase = kh + ((long long)b * NK + kt * KT) * HH;

  // qs: 4096 floats = 1024 x b128 chunks (rows contiguous, stride HH)
#pragma unroll
  for (int c = tid; c < (16 * HH) / 4; c += 256)
    async_copy_b128(&qs[c * 4], qbase + c * 4);
  // ks: 8192 floats = 2048 x b128 chunks; each chunk stays inside one row
#pragma unroll
  for (int c = tid; c < (KT * HH) / 4; c += 256) {
    const int row = c >> 6;                   // (c*4) / HH
    const int col = (c & 63) << 2;            // (c*4) % HH
    async_copy_b128(&ks[row * (HH + 1) + col], kbase + c * 4);
  }
  vs[tid] = v_w[tid];                         // blockDim == HH == 256
  wait_async_lds();
  __syncthreads();

  const int ki = tid & (KT - 1);              // 0..31 == lane id
  const int qi = tid >> 5;                    // 0..7  (wave id)
  const float* krow = ks + ki * (HH + 1);
  const float* q0   = qs + qi * HH;           // wave-uniform -> LDS broadcast
  const float* q1   = qs + (qi + 8) * HH;

  float a0 = 0.f, a1 = 0.f;
#pragma unroll 4
  for (int h = 0; h < HH; ++h) {
    const float kv = krow[h];
    const float vw = vs[h];
    a0 = fmaf(vw, fast_tanh(q0[h] + kv), a0);
    a1 = fmaf(vw, fast_tanh(q1[h] + kv), a1);
  }
  const long long obase = ((long long)b * NQ + qt * 16) * NK + kt * KT + ki;
  scores[obase + (long long)qi * NK]       = a0;
  scores[obase + (long long)(qi + 8) * NK] = a1;
}

// ---------------------------------------------------------------------------
// In-place masked softmax over the NK axis. One block per (b, q) row.
// ---------------------------------------------------------------------------
__global__ void masked_softmax(float* __restrict__ sc,
                               const int* __restrict__ valid_lens)
{
  const int row = blockIdx.x;                 // b*NQ + q
  const int b   = row / NQ;
  float* s = sc + (long long)row * NK;
  const int vl = valid_lens[b];

  __shared__ float red[256];
  const int tid = threadIdx.x;

  float v[4];
  float m = -3.402823466e38f;
#pragma unroll
  for (int i = 0; i < 4; ++i) {
    const int k = tid + (i << 8);
    const float x = (k < vl) ? s[k] : -1.0e6f;
    v[i] = x;
    m = fmaxf(m, x);
  }
  red[tid] = m; __syncthreads();
  for (int off = 128; off > 0; off >>= 1) {
    if (tid < off) red[tid] = fmaxf(red[tid], red[tid + off]);
    __syncthreads();
  }
  m = red[0]; __syncthreads();

  float sum = 0.f;
#pragma unroll
  for (int i = 0; i < 4; ++i) { v[i] = __expf(v[i] - m); sum += v[i]; }
  red[tid] = sum; __syncthreads();
  for (int off = 128; off > 0; off >>= 1) {
    if (tid < off) red[tid] += red[tid + off];
    __syncthreads();
  }
  const float inv = 1.0f / red[0];
#pragma unroll
  for (int i = 0; i < 4; ++i) s[tid + (i << 8)] = v[i] * inv;
}

// ---------------------------------------------------------------------------
extern "C" void kernel_launch(void* const* d_in, const int* in_sizes, int n_in,
                              void* d_out, int out_size, void* d_ws, size_t ws_size,
                              hipStream_t stream)
{
  const float* queries = (const float*)d_in[0];
  const float* keysp   = (const float*)d_in[1];
  const float* valuesp = (const float*)d_in[2];
  const int*   vlens   = (const int*)  d_in[3];
  const float* Wq      = (const float*)d_in[4];
  const float* Wk      = (const float*)d_in[5];
  const float* vw      = (const float*)d_in[6];
  float* out = (float*)d_out;

  float* qh = (float*)d_ws;                           // B*NQ*H  = 0.5 MB
  float* kh = qh + (size_t)BATCH * NQ * HH;           // B*NK*H  = 4 MB
  float* sc = kh + (size_t)BATCH * NK * HH;           // B*NQ*NK = 2 MB

  { // qh = queries @ W_q   (W shared across batch -> sB = 0)
    const int tiles = BATCH * (NQ / 32) * (HH / 32);  // 128
    wmma_gemm_f32<<<(tiles + 7) / 8, 256, 0, stream>>>(
        queries, Wq, qh, NQ, HH, DQ, DQ, HH, HH,
        (long long)NQ * DQ, 0LL, (long long)NQ * HH, tiles);
  }
  { // kh = keys @ W_k
    const int tiles = BATCH * (NK / 32) * (HH / 32);  // 1024
    wmma_gemm_f32<<<(tiles + 7) / 8, 256, 0, stream>>>(
        keysp, Wk, kh, NK, HH, DK, DK, HH, HH,
        (long long)NK * DK, 0LL, (long long)NK * HH, tiles);
  }

  additive_scores<<<BATCH * (NQ / 16) * (NK / KT), 256, 0, stream>>>(qh, kh, vw, sc);

  masked_softmax<<<BATCH * NQ, 256, 0, stream>>>(sc, vlens);

  { // out = weights @ values
    const int tiles = BATCH * (NQ / 32) * (DV / 32);  // 128
    wmma_gemm_f32<<<(tiles + 7) / 8, 256, 0, stream>>>(
        sc, valuesp, out, NQ, DV, NK, NK, DV, DV,
        (long long)NQ * NK, (long long)NK * DV, (long long)NQ * DV, tiles);
  }
}